// GraphAttentionLayer_multiEuli_84310208020843
// MI455X (gfx1250) — compile-verified
//
#include <hip/hip_runtime.h>
#include <math.h>

// ---------------------------------------------------------------------------
// GraphAttentionLayer_multiEuli for MI455X (gfx1250, wave32)
//
// Cost model: h@W reads 256MB of f32 W -> memory bound, ~11.5us at 23.3TB/s.
// Strategy: split-K WMMA (V_WMMA_F32_16X16X4_F32) GEMM with 8192 waves and
// 8x-unrolled load pipelining for HBM bandwidth; W loads are non-temporal
// (streamed once, > L2). Deterministic fixed-order reduction, then tiny
// distance/softmax/epilogue kernels.
// ---------------------------------------------------------------------------

#define IN_F   8192
#define OUT_F  8192
#define NTILES (OUT_F / 16)   // 512 column tiles
#define NCHUNK 16             // K split -> 8192 independent waves
#define KCHUNK (IN_F / NCHUNK)

typedef __attribute__((ext_vector_type(2))) float v2f;
typedef __attribute__((ext_vector_type(8))) float v8f;

// ---------------------------------------------------------------------------
// Kernel 1: hp_partial = h @ W  (split-K, WMMA f32 16x16x4)
// One wave per (column-tile, k-chunk). A rows 4..15 replicate rows 0..3
// (in-bounds; D rows 4..15 are discarded). Stores 4x16 partial per wave.
// ---------------------------------------------------------------------------
__global__ __launch_bounds__(256) void gat_gemm_wmma(
    const float* __restrict__ h, const float* __restrict__ W,
    float* __restrict__ part) {
  const int lane  = threadIdx.x & 31;
  const int wave  = threadIdx.x >> 5;
  const int gwave = blockIdx.x * 8 + wave;     // 0..8191
  const int tile  = gwave >> 4;                // 0..511
  const int chunk = gwave & (NCHUNK - 1);      // 0..15
  const int col0  = tile * 16;
  const int k0    = chunk * KCHUNK;

  const int mn   = lane & 15;                  // M (for A) / N (for B,D)
  const int kk   = (lane >> 4) * 2;            // K sub-offset: 0 or 2
  const float* __restrict__ hrow = h + (size_t)(mn & 3) * IN_F;
  const float* __restrict__ wcol = W + (size_t)col0 + mn;

  v8f acc = {0.f, 0.f, 0.f, 0.f, 0.f, 0.f, 0.f, 0.f};

#pragma unroll 8
  for (int k = k0; k < k0 + KCHUNK; k += 4) {
    // A fragment: lane L holds h[M=L&15][k+kk], h[...][k+kk+1]  (8B aligned,
    // hot in L1/L2 -> regular temporal)
    const float2 av = *reinterpret_cast<const float2*>(hrow + k + kk);
    v2f a; a.x = av.x; a.y = av.y;
    // B fragment: lane L holds W[k+kk][col0+N], W[k+kk+1][col0+N].
    // W is 256MB streamed exactly once -> non-temporal to spare L2.
    v2f b;
    b.x = __builtin_nontemporal_load(wcol + (size_t)(k + kk)     * OUT_F);
    b.y = __builtin_nontemporal_load(wcol + (size_t)(k + kk + 1) * OUT_F);
    // D = A*B + C  (16x16 f32 accumulator, 8 VGPRs)
    acc = __builtin_amdgcn_wmma_f32_16x16x4_f32(
        /*neg_a=*/false, a, /*neg_b=*/false, b,
        /*c_mod=*/(short)0, acc, /*reuse_a=*/false, /*reuse_b=*/false);
  }

  // D layout: VGPR r, lanes 0..15 hold M=r, N=lane. Keep rows 0..3 only.
  if (lane < 16) {
    float* p = part + ((size_t)tile * NCHUNK + chunk) * 64 + mn;
    p[0]  = acc[0];
    p[16] = acc[1];
    p[32] = acc[2];
    p[48] = acc[3];
  }
}

// ---------------------------------------------------------------------------
// Kernel 2: deterministic fixed-order reduction of the K-split partials.
// ---------------------------------------------------------------------------
__global__ __launch_bounds__(256) void gat_reduce(
    const float* __restrict__ part, float* __restrict__ hp) {
  const int idx = blockIdx.x * 256 + threadIdx.x;  // 0..32767
  const int col = idx & (OUT_F - 1);
  const int m   = idx >> 13;                       // 0..3
  const int tile = col >> 4;
  const int cn   = col & 15;
  const float* p = part + (size_t)tile * NCHUNK * 64 + m * 16 + cn;
  float s = 0.f;
#pragma unroll
  for (int c = 0; c < NCHUNK; ++c) s += p[c * 64];
  hp[(size_t)m * OUT_F + col] = s;
}

// ---------------------------------------------------------------------------
// Kernel 3: pairwise distances -> v01/v02/v12 -> adj-masked column softmax.
// Single block; writes att[9] to workspace and 3 scalars to d_out tail.
// Note leaky_relu is identity here (distances >= 0), matching the reference.
// ---------------------------------------------------------------------------
__global__ __launch_bounds__(256) void gat_dist(
    const float* __restrict__ hp, const int* __restrict__ adj,
    float* __restrict__ att, float* __restrict__ scal) {
  __shared__ float red[6][256];
  float s[6] = {0.f, 0.f, 0.f, 0.f, 0.f, 0.f};
  for (int k = threadIdx.x; k < OUT_F; k += 256) {
    const float h0 = hp[k];
    const float h1 = hp[OUT_F + k];
    const float h2 = hp[2 * OUT_F + k];
    const float h3 = hp[3 * OUT_F + k];
    float d;
    d = h0 - h1; s[0] += d * d;   // (0,1)
    d = h0 - h2; s[1] += d * d;   // (0,2)
    d = h0 - h3; s[2] += d * d;   // (0,3)
    d = h1 - h2; s[3] += d * d;   // (1,2)
    d = h1 - h3; s[4] += d * d;   // (1,3)
    d = h2 - h3; s[5] += d * d;   // (2,3)
  }
#pragma unroll
  for (int p = 0; p < 6; ++p) red[p][threadIdx.x] = s[p];
  __syncthreads();
  for (int off = 128; off >= 1; off >>= 1) {
    if ((int)threadIdx.x < off) {
#pragma unroll
      for (int p = 0; p < 6; ++p)
        red[p][threadIdx.x] += red[p][threadIdx.x + off];
    }
    __syncthreads();
  }
  if (threadIdx.x == 0) {
    float dd[6];
#pragma unroll
    for (int p = 0; p < 6; ++p) {
      const float sq = red[p][0];
      dd[p] = sq > 0.f ? sqrtf(sq) : 0.f;       // zero-safe cdist
    }
    const float d01 = dd[0], d02 = dd[1], d03 = dd[2];
    const float d12 = dd[3], d13 = dd[4], d23 = dd[5];
    const float v01 = d12 + 0.5f * (d01 + d02);
    const float v02 = d13 + 0.5f * (d01 + d03);
    const float v12 = d23 + 0.5f * (d02 + d03);
    float e[3][3] = {{0.f, v01, v02}, {v01, 0.f, v12}, {v02, v12, 0.f}};
    // adj mask + softmax along axis 0 (per column)
    for (int j = 0; j < 3; ++j) {
      float x[3], mx = -3.0e38f;
      for (int i = 0; i < 3; ++i) {
        x[i] = (adj[i * 3 + j] > 0) ? e[i][j] : -9.0e15f;
        mx = fmaxf(mx, x[i]);
      }
      float sum = 0.f;
      for (int i = 0; i < 3; ++i) { x[i] = __expf(x[i] - mx); sum += x[i]; }
      const float inv = 1.0f / sum;
      for (int i = 0; i < 3; ++i) att[i * 3 + j] = x[i] * inv;
    }
    scal[0] = v01;  // face_Rhand
    scal[1] = v02;  // face_Lhand
    scal[2] = v12;  // Rhand_Lhand
  }
}

// ---------------------------------------------------------------------------
// Kernel 4: epilogue. Row 0 = hp[0]; rows 1..3 = attention @ hp[1:4].
// ---------------------------------------------------------------------------
__global__ __launch_bounds__(256) void gat_out(
    const float* __restrict__ hp, const float* __restrict__ att,
    float* __restrict__ out) {
  const int idx = blockIdx.x * 256 + threadIdx.x;  // 0..32767
  const int col = idx & (OUT_F - 1);
  const int row = idx >> 13;
  float v;
  if (row == 0) {
    v = hp[col];
  } else {
    const int i = row - 1;
    v = att[i * 3 + 0] * hp[(size_t)1 * OUT_F + col] +
        att[i * 3 + 1] * hp[(size_t)2 * OUT_F + col] +
        att[i * 3 + 2] * hp[(size_t)3 * OUT_F + col];
  }
  out[idx] = v;
}

// ---------------------------------------------------------------------------
extern "C" void kernel_launch(void* const* d_in, const int* in_sizes, int n_in,
                              void* d_out, int out_size, void* d_ws, size_t ws_size,
                              hipStream_t stream) {
  const float* h   = (const float*)d_in[0];   // [4, 8192] f32
  const int*   adj = (const int*)d_in[1];     // [3, 3] int32
  const float* W   = (const float*)d_in[2];   // [8192, 8192] f32
  float* out = (float*)d_out;                 // 4*8192 h_prime + 3 scalars
  float* ws  = (float*)d_ws;

  float* part = ws;                                   // 512*16*64 = 524288 f32
  float* hp   = ws + (size_t)NTILES * NCHUNK * 64;    // 32768 f32
  float* att  = hp + (size_t)4 * OUT_F;               // 9 f32

  gat_gemm_wmma<<<(NTILES * NCHUNK) / 8, 256, 0, stream>>>(h, W, part);
  gat_reduce<<<(4 * OUT_F) / 256, 256, 0, stream>>>(part, hp);
  gat_dist<<<1, 256, 0, stream>>>(hp, adj, att, out + (size_t)4 * OUT_F);
  gat_out<<<(4 * OUT_F) / 256, 256, 0, stream>>>(hp, att, out);
}